// Embedding_71760313581990
// MI455X (gfx1250) — compile-verified
//
#include <hip/hip_runtime.h>

// Embedding gather: out[row, :] = weight[token_ids[row], :]
//   token_ids: int32 [4*4096]          (d_in[0])
//   weight   : float32 [50257, 1024]   (d_in[1])
//   out      : float32 [16384, 1024]   (d_out)
//
// Pure bandwidth op: ~134 MB @ 23.3 TB/s ≈ 5.8 us. One 256-thread block per
// row; each thread moves one 16-byte chunk (256 * 16B = 4KB = one row) via the
// CDNA5 async global->LDS->global DMA path (ASYNCcnt-tracked, bypasses VGPRs).

#define D_MODEL 1024

typedef int v4i __attribute__((vector_size(16)));

// Device-pass feature probe (host pass must not see amdgcn builtins).
#if defined(__AMDGCN__) && \
    __has_builtin(__builtin_amdgcn_global_load_async_to_lds_b128) && \
    __has_builtin(__builtin_amdgcn_global_store_async_from_lds_b128)
#define EMB_USE_ASYNC_LDS 1
#else
#define EMB_USE_ASYNC_LDS 0
#endif

#if EMB_USE_ASYNC_LDS
typedef __attribute__((address_space(1))) v4i* gv4i_ptr;  // global
typedef __attribute__((address_space(3))) v4i* lv4i_ptr;  // LDS
#endif

__global__ __launch_bounds__(256) void embed_row_gather(
    const int* __restrict__ tok,
    const float* __restrict__ w,
    float* __restrict__ out) {
  const unsigned row = blockIdx.x;        // 0 .. 16383
  const unsigned i   = threadIdx.x;       // 0 .. 255, 16 bytes each
  const int t = tok[row];                 // uniform per block -> scalar load
  const size_t src_off = (size_t)t * D_MODEL;
  const size_t dst_off = (size_t)row * D_MODEL;

#if EMB_USE_ASYNC_LDS
  // Stage the 4KB row through LDS with async DMA ops. Each lane's async load
  // writes LDS bytes [16*lane, 16*lane+16); the same lane's async store reads
  // them back, so a per-wave ASYNCcnt==0 wait is the only ordering required.
  __shared__ float buf[D_MODEL];

  gv4i_ptr gsrc = (gv4i_ptr)(w + src_off + (size_t)i * 4u);   // cast drops const
  lv4i_ptr lbuf = (lv4i_ptr)(buf + (size_t)i * 4u);
  gv4i_ptr gdst = (gv4i_ptr)(out + dst_off + (size_t)i * 4u);

  __builtin_amdgcn_global_load_async_to_lds_b128(gsrc, lbuf, /*offset=*/0, /*cpol=*/0);

#if __has_builtin(__builtin_amdgcn_s_wait_asynccnt)
  __builtin_amdgcn_s_wait_asynccnt(0);
#else
  asm volatile("s_wait_asynccnt 0x0" ::: "memory");
#endif

  __builtin_amdgcn_global_store_async_from_lds_b128(gdst, lbuf, /*offset=*/0, /*cpol=*/0);
  // Implicit wait-idle at s_endpgm drains the outstanding async store.
#else
  // Fallback: direct coalesced 128-bit copy with a non-temporal store
  // (output is write-once; keep it from evicting table rows from L2).
  // Plain vector types so __builtin_nontemporal_* accepts them on both passes.
  const v4i v = ((const v4i*)(w + src_off))[i];
  __builtin_nontemporal_store(v, ((v4i*)(out + dst_off)) + i);
#endif
}

extern "C" void kernel_launch(void* const* d_in, const int* in_sizes, int n_in,
                              void* d_out, int out_size, void* d_ws, size_t ws_size,
                              hipStream_t stream) {
  const int*   tok = (const int*)d_in[0];     // int32 token ids, flat [B*S]
  const float* w   = (const float*)d_in[1];   // float32 [VOCAB, D_MODEL]
  float*       out = (float*)d_out;           // float32 [B*S, D_MODEL]

  const int rows = in_sizes[0];               // 16384
  embed_row_gather<<<rows, 256, 0, stream>>>(tok, w, out);
}